// DeformAttn_67525475827771
// MI455X (gfx1250) — compile-verified
//
#include <hip/hip_runtime.h>
#include <hip/hip_bf16.h>
#include <math.h>

// ---------------------------------------------------------------------------
// Deformable attention for MI455X (gfx1250, wave32).
//
// Key layout decision: the bilinear-gather stage dominates; with channel-major
// kv each 16-channel corner read touches 16 cache lines (4B useful each). We
// therefore keep all attention-facing tensors PIXEL-MAJOR:
//   qp_t[b][n][192], kv_t[b*2+c][n][384], o_t[b][n][192], h_t[b][n][384]
// so a corner gather is 64 contiguous bytes (4x global_load_b128).
// GEMMs use V_WMMA_F32_16X16X4_F32 (fp32 in/out -- the problem is memory
// bound, ~7.5 GFLOP vs ~75MB @ 23.3 TB/s, so fp32 costs nothing and keeps
// full precision). Final channel-major output via LDS-tiled transpose.
// ---------------------------------------------------------------------------

typedef __attribute__((ext_vector_type(2))) float v2f;
typedef __attribute__((ext_vector_type(8))) float v8f;

#define HWN   2304      // H*W
#define HH    48
#define WW    48
#define CCH   192       // C
#define NHEAD 12
#define DIMH  16
#define KPTS  9
#define NCLIP 2

__device__ __forceinline__ float dot4(const float4& a, const float4& b) {
    return a.x * b.x + a.y * b.y + a.z * b.z + a.w * b.w;
}
__device__ __forceinline__ void fma4(float4& acc, float w, const float4& v) {
    acc.x += w * v.x; acc.y += w * v.y; acc.z += w * v.z; acc.w += w * v.w;
}

// ---------------------------------------------------------------------------
// Projection GEMM (channel-major input, TRANSPOSED output):
//   Yt[n][colOff + m] = sum_k W[m,k] * X[k,n] + bias[m]
// A = W (row m, contiguous k -> float2), B = X (row k, col n).
// Block = 8 waves (4M x 2N); wave tile 16 x 64. Batched over blockIdx.z.
// ---------------------------------------------------------------------------
__global__ __launch_bounds__(256) void wmma_gemm_proj(
    const float* __restrict__ Wm, const float* __restrict__ X,
    const float* __restrict__ bias, float* __restrict__ Yt,
    int K, int N, int ldOut, int colOff, long sX, long sYt)
{
    const int lane  = threadIdx.x & 31;
    const int wave  = threadIdx.x >> 5;
    const int mBase = blockIdx.y * 64 + (wave & 3) * 16;
    const int nBase = blockIdx.x * 128 + (wave >> 2) * 64;

    X  += (long)blockIdx.z * sX;
    Yt += (long)blockIdx.z * sYt;

    const int mLane = lane & 15;
    const int kLane = lane >> 4;

    v8f acc[4] = {v8f{}, v8f{}, v8f{}, v8f{}};

    const float* wrow  = Wm + (long)(mBase + mLane) * K + 2 * kLane;
    const float* xcol0 = X + (long)(2 * kLane) * N + nBase + mLane;

    for (int k0 = 0; k0 < K; k0 += 4) {
        v2f a = *(const v2f*)(wrow + k0);
        const float* xc = xcol0 + (long)k0 * N;
        v2f b0, b1, b2, b3;
        b0.x = xc[0];   b0.y = xc[N];
        b1.x = xc[16];  b1.y = xc[N + 16];
        b2.x = xc[32];  b2.y = xc[N + 32];
        b3.x = xc[48];  b3.y = xc[N + 48];
        acc[0] = __builtin_amdgcn_wmma_f32_16x16x4_f32(false, a, false, b0, (short)0, acc[0], false, false);
        acc[1] = __builtin_amdgcn_wmma_f32_16x16x4_f32(false, a, false, b1, (short)0, acc[1], false, false);
        acc[2] = __builtin_amdgcn_wmma_f32_16x16x4_f32(false, a, false, b2, (short)0, acc[2], false, false);
        acc[3] = __builtin_amdgcn_wmma_f32_16x16x4_f32(false, a, false, b3, (short)0, acc[3], false, false);
    }

    // C/D layout: VGPR r -> row (mBase + r + kLane*8), col (nBase + j*16 + mLane).
    // Transposed store: lane writes 8 contiguous channels of one pixel row.
    #pragma unroll
    for (int j = 0; j < 4; ++j) {
        const int nCol = nBase + j * 16 + mLane;
        float* yrow = Yt + (long)nCol * ldOut + colOff + mBase + kLane * 8;
        const float* brow = bias + mBase + kLane * 8;
        #pragma unroll
        for (int r = 0; r < 8; ++r)
            yrow[r] = acc[j][r] + brow[r];
    }
}

// ---------------------------------------------------------------------------
// NT GEMM on pixel-major activations:
//   Y[pix][oc] = sum_k A[pix][k] * Wgt[oc][k] + bias[oc]   (+gelu / +residual)
// Both fragments are K-contiguous float2 loads. Block = 8 waves along M
// (pixels, 128) x 1 along N; wave tile 16 x 64 (4 accumulators).
// ---------------------------------------------------------------------------
__global__ __launch_bounds__(256) void wmma_gemm_nt(
    const float* __restrict__ A, const float* __restrict__ Wgt,
    const float* __restrict__ bias, const float* __restrict__ R,
    float* __restrict__ Y, int K, int ldY, int ldR, int act_gelu)
{
    const int lane  = threadIdx.x & 31;
    const int wave  = threadIdx.x >> 5;
    const int mBase = blockIdx.y * 128 + wave * 16;   // pixel rows
    const int nBase = blockIdx.x * 64;                // output channels

    const int mLane = lane & 15;
    const int kLane = lane >> 4;

    v8f acc[4] = {v8f{}, v8f{}, v8f{}, v8f{}};

    const float* arow = A + (long)(mBase + mLane) * K + 2 * kLane;
    const float* w0   = Wgt + (long)(nBase + mLane) * K + 2 * kLane;
    const float* w1r  = w0 + (long)16 * K;
    const float* w2r  = w0 + (long)32 * K;
    const float* w3r  = w0 + (long)48 * K;

    for (int k0 = 0; k0 < K; k0 += 4) {
        v2f a  = *(const v2f*)(arow + k0);
        v2f b0 = *(const v2f*)(w0 + k0);
        v2f b1 = *(const v2f*)(w1r + k0);
        v2f b2 = *(const v2f*)(w2r + k0);
        v2f b3 = *(const v2f*)(w3r + k0);
        acc[0] = __builtin_amdgcn_wmma_f32_16x16x4_f32(false, a, false, b0, (short)0, acc[0], false, false);
        acc[1] = __builtin_amdgcn_wmma_f32_16x16x4_f32(false, a, false, b1, (short)0, acc[1], false, false);
        acc[2] = __builtin_amdgcn_wmma_f32_16x16x4_f32(false, a, false, b2, (short)0, acc[2], false, false);
        acc[3] = __builtin_amdgcn_wmma_f32_16x16x4_f32(false, a, false, b3, (short)0, acc[3], false, false);
    }

    #pragma unroll
    for (int j = 0; j < 4; ++j) {
        const int oc = nBase + j * 16 + mLane;
        const float bval = bias[oc];
        #pragma unroll
        for (int r = 0; r < 8; ++r) {
            const int pix = mBase + r + kLane * 8;
            float v = acc[j][r] + bval;
            if (act_gelu) v = 0.5f * v * (1.0f + erff(v * 0.70710678118654752f));
            if (R) v += R[(long)pix * ldR + oc];
            Y[(long)pix * ldY + oc] = v;   // lanes 0-15 contiguous oc -> 64B coalesced
        }
    }
}

// ---------------------------------------------------------------------------
// Deformable attention. One thread per (b, head, pixel), pixel-major tensors.
// kv_t channel f = half*192 + head*16 + d; deform group of a (head,half) is
// g = f/32 -> gk = head/2 (K-samples), gv = 6 + head/2 (V-samples).
// Each bilinear corner = 4x float4 contiguous loads (one cache line).
// ---------------------------------------------------------------------------
__global__ __launch_bounds__(256) void deform_attn_kernel(
    const float* __restrict__ qp, const float* __restrict__ kv,
    const float* __restrict__ off, float* __restrict__ o)
{
    const int n = blockIdx.x * 256 + threadIdx.x;   // pixel (HW = 9*256)
    const int h = blockIdx.y;
    const int b = blockIdx.z;
    const int py0 = n / WW, px0 = n % WW;

    // qh = q * dim^-0.5 (0.25), 16 contiguous floats
    const float4* qv = (const float4*)(qp + ((long)b * HWN + n) * CCH + h * DIMH);
    float4 qh[4] = {qv[0], qv[1], qv[2], qv[3]};
    #pragma unroll
    for (int i = 0; i < 4; ++i) {
        qh[i].x *= 0.25f; qh[i].y *= 0.25f; qh[i].z *= 0.25f; qh[i].w *= 0.25f;
    }

    const int gk = h >> 1;
    const int gv = 6 + (h >> 1);

    float sc[NCLIP * KPTS];

    // ---- pass 1: logits (gather K-channels) ----
    for (int c = 0; c < NCLIP; ++c) {
        const float* kvc  = kv + ((long)(b * NCLIP + c) * HWN) * 384 + h * DIMH;
        const float* offp = off + (long)(b * NCLIP + c) * 216 * HWN + n;
        for (int k = 0; k < KPTS; ++k) {
            const float oy = offp[(long)((gk * KPTS + k) * 2 + 0) * HWN];
            const float ox = offp[(long)((gk * KPTS + k) * 2 + 1) * HWN];
            const float py = (float)(py0 - 1 + k / 3) + oy;
            const float px = (float)(px0 - 1 + k % 3) + ox;
            const float fy = floorf(py), fx = floorf(px);
            const int   y0 = (int)fy,   x0 = (int)fx;
            const float wy = py - fy,   wx = px - fx;
            float s = 0.0f;
            #pragma unroll
            for (int corner = 0; corner < 4; ++corner) {
                const int   yi  = y0 + (corner >> 1);
                const int   xi  = x0 + (corner & 1);
                const float wgt = ((corner >> 1) ? wy : 1.0f - wy) *
                                  ((corner & 1)  ? wx : 1.0f - wx);
                if (yi >= 0 && yi < HH && xi >= 0 && xi < WW) {
                    const long idx = yi * WW + xi;
                    const float4* p = (const float4*)(kvc + idx * 384);
                    s += wgt * (dot4(qh[0], p[0]) + dot4(qh[1], p[1]) +
                                dot4(qh[2], p[2]) + dot4(qh[3], p[3]));
                }
            }
            sc[c * KPTS + k] = s;
        }
    }

    // ---- softmax over 18 ----
    float mx = sc[0];
    #pragma unroll
    for (int i = 1; i < NCLIP * KPTS; ++i) mx = fmaxf(mx, sc[i]);
    float sum = 0.0f;
    #pragma unroll
    for (int i = 0; i < NCLIP * KPTS; ++i) { sc[i] = expf(sc[i] - mx); sum += sc[i]; }
    const float inv = 1.0f / sum;

    // ---- pass 2: weighted V gather ----
    float4 outv[4] = {};
    for (int c = 0; c < NCLIP; ++c) {
        const float* kvvc = kv + ((long)(b * NCLIP + c) * HWN) * 384 + 192 + h * DIMH;
        const float* offp = off + (long)(b * NCLIP + c) * 216 * HWN + n;
        for (int k = 0; k < KPTS; ++k) {
            const float attn = sc[c * KPTS + k] * inv;
            const float oy = offp[(long)((gv * KPTS + k) * 2 + 0) * HWN];
            const float ox = offp[(long)((gv * KPTS + k) * 2 + 1) * HWN];
            const float py = (float)(py0 - 1 + k / 3) + oy;
            const float px = (float)(px0 - 1 + k % 3) + ox;
            const float fy = floorf(py), fx = floorf(px);
            const int   y0 = (int)fy,   x0 = (int)fx;
            const float wy = py - fy,   wx = px - fx;
            #pragma unroll
            for (int corner = 0; corner < 4; ++corner) {
                const int   yi  = y0 + (corner >> 1);
                const int   xi  = x0 + (corner & 1);
                const float wgt = ((corner >> 1) ? wy : 1.0f - wy) *
                                  ((corner & 1)  ? wx : 1.0f - wx);
                if (yi >= 0 && yi < HH && xi >= 0 && xi < WW) {
                    const long idx = yi * WW + xi;
                    const float4* p = (const float4*)(kvvc + idx * 384);
                    const float w2 = attn * wgt;
                    fma4(outv[0], w2, p[0]); fma4(outv[1], w2, p[1]);
                    fma4(outv[2], w2, p[2]); fma4(outv[3], w2, p[3]);
                }
            }
        }
    }

    float4* op = (float4*)(o + ((long)b * HWN + n) * CCH + h * DIMH);
    #pragma unroll
    for (int i = 0; i < 4; ++i) op[i] = outv[i];
}

// ---------------------------------------------------------------------------
// LDS-tiled transpose: pixel-major [2304][192] -> channel-major [192][2304].
// 32x32 tile (+1 pad, no bank conflicts), coalesced loads and stores.
// ---------------------------------------------------------------------------
__global__ __launch_bounds__(256) void transpose_kernel(
    const float* __restrict__ in, float* __restrict__ out)
{
    __shared__ float tile[32][33];
    const int b = blockIdx.z;
    in  += (long)b * HWN * CCH;
    out += (long)b * CCH * HWN;
    const int p0 = blockIdx.x * 32;     // pixel tile
    const int c0 = blockIdx.y * 32;     // channel tile
    const int tx = threadIdx.x & 31;
    const int ty = threadIdx.x >> 5;    // 0..7

    #pragma unroll
    for (int i = 0; i < 4; ++i)
        tile[ty + i * 8][tx] = in[(long)(p0 + ty + i * 8) * CCH + c0 + tx];
    __syncthreads();
    #pragma unroll
    for (int i = 0; i < 4; ++i)
        out[(long)(c0 + ty + i * 8) * HWN + p0 + tx] = tile[tx][ty + i * 8];
}

// ---------------------------------------------------------------------------
// Host side. Workspace (floats), 10,616,832 total = 42.5 MB:
//   qp_t : [0,        1769472)
//   kv_t : [1769472,  8847360)
//   o_t  : [8847360, 10616832)
//   h_t  : aliases [0, 3538944)        (qp_t + kv_t head, dead after attn)
//   out_t: aliases [3538944, 5308416)  (inside dead kv_t region)
// ---------------------------------------------------------------------------
extern "C" void kernel_launch(void* const* d_in, const int* in_sizes, int n_in,
                              void* d_out, int out_size, void* d_ws, size_t ws_size,
                              hipStream_t stream) {
    const float* q   = (const float*)d_in[0];
    const float* k   = (const float*)d_in[1];
    const float* v   = (const float*)d_in[2];
    const float* off = (const float*)d_in[3];
    const float* wq  = (const float*)d_in[4];
    const float* bq  = (const float*)d_in[5];
    const float* wk  = (const float*)d_in[6];
    const float* bk  = (const float*)d_in[7];
    const float* wv  = (const float*)d_in[8];
    const float* bv  = (const float*)d_in[9];
    const float* w1  = (const float*)d_in[10];
    const float* b1  = (const float*)d_in[11];
    const float* w2  = (const float*)d_in[12];
    const float* b2  = (const float*)d_in[13];
    float* out = (float*)d_out;

    float* ws = (float*)d_ws;
    const long IMG = (long)CCH * HWN;       // 442368 floats
    float* qp_t  = ws;
    float* kv_t  = ws + 4 * IMG;            // 8 * 2*IMG
    float* o_t   = kv_t + 16 * IMG;
    float* h_t   = ws;                      // 8*IMG, reuses qp_t + kv_t head
    float* out_t = ws + 8 * IMG;            // 4*IMG, inside dead kv_t region

    dim3 blk(256);

    // q/k/v projections (channel-major in, pixel-major out)
    wmma_gemm_proj<<<dim3(18, 3, 4), blk, 0, stream>>>(
        wq, q, bq, qp_t, 192, HWN, 192, 0, IMG, IMG);
    wmma_gemm_proj<<<dim3(18, 3, 8), blk, 0, stream>>>(
        wk, k, bk, kv_t, 192, HWN, 384, 0, IMG, 2 * IMG);
    wmma_gemm_proj<<<dim3(18, 3, 8), blk, 0, stream>>>(
        wv, v, bv, kv_t, 192, HWN, 384, 192, IMG, 2 * IMG);

    // deformable attention -> o_t (pixel-major)
    deform_attn_kernel<<<dim3(HWN / 256, NHEAD, 4), blk, 0, stream>>>(qp_t, kv_t, off, o_t);

    // FFN1: h = gelu(o . w1^T + b1), M=9216 pixels, N=384, K=192
    wmma_gemm_nt<<<dim3(6, 72), blk, 0, stream>>>(
        o_t, w1, b1, nullptr, h_t, 192, 384, 0, 1);

    // FFN2 + residual: out_t = o + (h . w2^T + b2), N=192, K=384
    wmma_gemm_nt<<<dim3(3, 72), blk, 0, stream>>>(
        h_t, w2, b2, o_t, out_t, 384, 192, 192, 0);

    // pixel-major -> channel-major final output
    transpose_kernel<<<dim3(HWN / 32, CCH / 32, 4), blk, 0, stream>>>(out_t, out);
}